// Model_87333864997450
// MI455X (gfx1250) — compile-verified
//
#include <hip/hip_runtime.h>
#include <stdint.h>

// -------------------------------------------------------------------------
// Ragged pos/seq_lens kernel for gfx1250 (MI455X).
// Bandwidth-bound: 64 MB of int32 streaming stores => ~2.8 us at 23.3 TB/s.
// No matrix math exists in this op, so no WMMA; optimization = coalesced
// 128-bit NT stores + balanced fixed-size chunking + block-uniform bracketing
// binary search instead of per-token searchsorted.
// -------------------------------------------------------------------------

typedef int vint4 __attribute__((ext_vector_type(4)));  // true clang vector

namespace {
constexpr int TPB          = 256;              // 8 waves (wave32) per block
constexpr int GROUPS       = 2;                // 128-bit groups per thread
constexpr int TOK_PER_BLK  = TPB * GROUPS * 4; // 2048 tokens per block
}

// Largest r in [lo, hi] with qsl[r] <= t.  Caller guarantees qsl[lo] <= t.
// Equivalent to searchsorted(qsl, t, side='right') - 1 clipped to [0, hi].
__device__ __forceinline__ int find_req(const int* __restrict__ qsl,
                                        int t, int lo, int hi) {
  while (lo < hi) {
    int mid = (lo + hi + 1) >> 1;
    if (qsl[mid] <= t) lo = mid;
    else               hi = mid - 1;
  }
  return lo;
}

__device__ __forceinline__ void nt_store4(int* p, int a, int b, int c, int d) {
  vint4 v = {a, b, c, d};
  __builtin_nontemporal_store(v, reinterpret_cast<vint4*>(p));
}

// Kernel 1: seq_lens (16384 elems) + materialize base[r] = nct[idx_mapping[r]]
// into workspace so the token kernel has a single gather level.
__global__ __launch_bounds__(TPB)
void seqlens_base_kernel(const int* __restrict__ idx_mapping,
                         const int* __restrict__ qsl,
                         const int* __restrict__ nct,
                         int* __restrict__ seq_out,
                         int* __restrict__ base_ws,
                         int num_reqs, int max_num_reqs) {
  int i = blockIdx.x * TPB + threadIdx.x;
  if (i >= max_num_reqs) return;
  if (i < num_reqs) {
    int b = nct[idx_mapping[i]];
    base_ws[i] = b;
    seq_out[i] = b + (qsl[i + 1] - qsl[i]);
  } else {
    seq_out[i] = 0;
  }
}

// Kernel 2: pos[t] = base[req(t)] + (t - qsl[req(t)]) over fixed 2048-token
// chunks. Block-uniform binary search brackets the chunk's request range;
// boundary-free chunks degenerate to a pure streaming store.
__global__ __launch_bounds__(TPB)
void pos_kernel(const int* __restrict__ qsl,
                const int* __restrict__ base_ws,
                int* __restrict__ pos_out,
                int num_reqs, int num_tokens) {
  const int c0       = (int)blockIdx.x * TOK_PER_BLK;
  const int last_req = num_reqs - 1;
  int chunk_last = c0 + TOK_PER_BLK - 1;
  if (chunk_last >= num_tokens) chunk_last = num_tokens - 1;

  // Block-uniform bracket (qsl[0] == 0 <= c0 always holds).
  const int r_lo = find_req(qsl, c0, 0, last_req);
  const int r_hi = find_req(qsl, chunk_last, r_lo, last_req);

  // gfx1250 prefetch path (global_prefetch_b8) for the descriptor window.
  __builtin_prefetch((const void*)(qsl + r_lo), 0, 0);
  __builtin_prefetch((const void*)(base_ws + r_lo), 0, 0);

  if (r_lo == r_hi) {
    // Whole chunk owned by one request: pos[t] = addend + t. Pure stream.
    const int addend = base_ws[r_lo] - qsl[r_lo];
#pragma unroll
    for (int g = 0; g < GROUPS; ++g) {
      int t0 = c0 + (g * TPB + (int)threadIdx.x) * 4;
      if (t0 + 3 < num_tokens) {
        nt_store4(pos_out + t0, addend + t0, addend + t0 + 1,
                  addend + t0 + 2, addend + t0 + 3);
      } else {
        for (int k = 0; k < 4; ++k) {
          int t = t0 + k;
          if (t >= num_tokens) break;
          pos_out[t] = addend + t;
        }
      }
    }
    return;
  }

  // Chunk spans multiple requests: one small bounded search per int4 group.
#pragma unroll
  for (int g = 0; g < GROUPS; ++g) {
    int t0 = c0 + (g * TPB + (int)threadIdx.x) * 4;
    if (t0 >= num_tokens) continue;
    int req   = find_req(qsl, t0, r_lo, r_hi);
    int bound = qsl[req + 1];                 // req + 1 <= num_reqs, in range
    if (t0 + 3 < bound && t0 + 3 < num_tokens) {
      int addend = base_ws[req] - qsl[req];
      nt_store4(pos_out + t0, addend + t0, addend + t0 + 1,
                addend + t0 + 2, addend + t0 + 3);
    } else {
      // Group crosses a request boundary (rare): per-element resolve.
      for (int k = 0; k < 4; ++k) {
        int t = t0 + k;
        if (t >= num_tokens) break;
        int r = find_req(qsl, t, req, r_hi);
        pos_out[t] = base_ws[r] + (t - qsl[r]);
      }
    }
  }
}

extern "C" void kernel_launch(void* const* d_in, const int* in_sizes, int n_in,
                              void* d_out, int out_size, void* d_ws, size_t ws_size,
                              hipStream_t stream) {
  // setup_inputs() order: idx_mapping, query_start_loc, num_computed_tokens,
  //                       pos (output-shaped), seq_lens (output-shaped)
  const int* idx_mapping = (const int*)d_in[0];
  const int* qsl         = (const int*)d_in[1];
  const int* nct         = (const int*)d_in[2];

  const int num_reqs     = in_sizes[0];   // 8192
  const int num_tokens   = in_sizes[3];   // 16,777,216
  const int max_num_reqs = in_sizes[4];   // 16384

  // d_out = [pos (num_tokens int32) | seq_lens (max_num_reqs int32)], stored
  // as raw int32 bit patterns in the 4-byte-element output buffer.
  int* pos_out = (int*)d_out;
  int* seq_out = (int*)d_out + num_tokens;
  int* base_ws = (int*)d_ws;              // num_reqs int32 of scratch

  const int blocks1 = (max_num_reqs + TPB - 1) / TPB;
  seqlens_base_kernel<<<blocks1, TPB, 0, stream>>>(
      idx_mapping, qsl, nct, seq_out, base_ws, num_reqs, max_num_reqs);

  const int blocks2 = (num_tokens + TOK_PER_BLK - 1) / TOK_PER_BLK;
  pos_kernel<<<blocks2, TPB, 0, stream>>>(
      qsl, base_ws, pos_out, num_reqs, num_tokens);
}